// GNNLayer_28003186770155
// MI455X (gfx1250) — compile-verified
//
#include <hip/hip_runtime.h>
#include <cstdint>

// ---------------------------------------------------------------------------
// GNN layer for MI455X (gfx1250, wave32):
//   h   = x @ W^T + b   -> fp32 WMMA (V_WMMA_F32_16X16X4_F32),
//                          x-stripe staged via GLOBAL_LOAD_ASYNC_TO_LDS_B128
//   out = segment_sum(vals * h[cols], rows) -> wave-per-edge native
//                          GLOBAL_ATOMIC_ADD_F32 (no-return, L2-resident)
//
// Roofline: GEMM is 3.3 GFLOP / ~100MB (trivial). Aggregation moves ~819MB of
// random h-row gathers + ~205M f32 atomics, but h and out (51.2MB each) fit
// in the 192MB L2, so plain L2-side atomics are bandwidth-optimal — no edge
// sorting needed.
// ---------------------------------------------------------------------------

#define DIM 128      // DIM_IN == DIM_OUT
#define PITCH 132    // LDS row pitch in floats: 528B = 16B-aligned rows (b128
                     // async stores OK) and 132 mod 64 = 4 -> the 16 lanes of
                     // an A-fragment ds_load_b64 hit 16 distinct banks.

typedef __attribute__((ext_vector_type(2))) float v2f;
typedef __attribute__((ext_vector_type(8))) float v8f;

// ----------------------------- zero output --------------------------------
__global__ void __launch_bounds__(256) gnn_zero_out(float* __restrict__ out, int n) {
    int i = blockIdx.x * blockDim.x + threadIdx.x;
    if (i < n) out[i] = 0.0f;
}

// -------------------------- transpose W -> Wt ------------------------------
// wt[k*DIM + n] = w[n*DIM + k] so B-fragment loads in the GEMM are coalesced.
__global__ void __launch_bounds__(256) gnn_transpose_w(const float* __restrict__ w,
                                                       float* __restrict__ wt) {
    int i = blockIdx.x * blockDim.x + threadIdx.x;   // 0 .. 128*128-1
    if (i < DIM * DIM) {
        int n = i >> 7;
        int k = i & (DIM - 1);
        wt[k * DIM + n] = w[i];
    }
}

// ------------------------------ WMMA GEMM ----------------------------------
// Block = 256 threads = 8 waves. Each block computes a 16-row stripe of h;
// wave m computes the 16x16 tile at columns [16m, 16m+16).
// K=128 consumed in 32 steps of V_WMMA_F32_16X16X4_F32.
//
// Fragment layouts (ISA 7.12.2, wave32):
//   A (16x4 f32):  lane L<16 holds M=L;    vgpr r holds K = r + 2*(L>=16)
//   B (4x16 f32):  lane L holds N = L&15;  vgpr r holds K = r + 2*(L>=16)
//   C/D (16x16):   lane L holds N = L&15;  vgpr i holds M = i + 8*(L>=16)
__global__ void __launch_bounds__(256) gnn_gemm_bias(const float* __restrict__ x,
                                                     const float* __restrict__ wt,   // [k][n]
                                                     const float* __restrict__ bias,
                                                     float* __restrict__ h,
                                                     int nNodes) {
    __shared__ float xs[16][PITCH];

    const int  rowBase = blockIdx.x * 16;
    const bool full    = (rowBase + 16 <= nNodes);

    if (full) {
        // Async DMA of the 16x128 f32 stripe into LDS: 512 b128 chunks,
        // 2 per thread, tracked by ASYNCcnt (frees VGPRs/issue during fill).
        const uint32_t ldsBase = (uint32_t)(uintptr_t)&xs[0][0];
        for (int c = threadIdx.x; c < 512; c += 256) {
            const int row  = c >> 5;            // 0..15
            const int col4 = (c & 31) << 2;     // 0,4,...,124 (floats)
            const uint32_t la = ldsBase + (uint32_t)(row * PITCH + col4) * 4u;
            const uint64_t ga = (uint64_t)(uintptr_t)
                                (x + (size_t)(rowBase + row) * DIM + col4);
            asm volatile("global_load_async_to_lds_b128 %0, %1, off"
                         :: "v"(la), "v"(ga) : "memory");
        }
        asm volatile("s_wait_asynccnt 0x0" ::: "memory");
    } else {
        // Tail block (never taken when nNodes % 16 == 0): guarded fill.
        for (int i = threadIdx.x; i < 16 * DIM; i += 256) {
            int r = i >> 7, cc = i & (DIM - 1);
            int gr = rowBase + r;
            xs[r][cc] = (gr < nNodes) ? x[(size_t)gr * DIM + cc] : 0.0f;
        }
    }
    __syncthreads();

    const int wave    = threadIdx.x >> 5;   // 0..7 -> column tile
    const int lane    = threadIdx.x & 31;
    const int colBase = wave * 16;
    const int half    = lane >> 4;          // 0 | 1
    const int l15     = lane & 15;

    v8f c = {};

#pragma unroll 8
    for (int k = 0; k < DIM; k += 4) {
        const int kk = k + 2 * half;

        // A fragment: two consecutive K values of row l15 (ds_load_b64).
        const float2 a2 = *(const float2*)&xs[l15][kk];
        v2f a; a.x = a2.x; a.y = a2.y;

        // B fragment: Wt rows kk, kk+1 at column colBase+l15 (coalesced).
        v2f b;
        b.x = wt[(size_t)(kk + 0) * DIM + colBase + l15];
        b.y = wt[(size_t)(kk + 1) * DIM + colBase + l15];

        c = __builtin_amdgcn_wmma_f32_16x16x4_f32(
                /*neg_a=*/false, a, /*neg_b=*/false, b,
                /*c_mod=*/(short)0, c, /*reuse_a=*/false, /*reuse_b=*/false);
    }

    const float bv = bias[colBase + l15];
    float* hp = h + (size_t)(rowBase + 8 * half) * DIM + colBase + l15;

    if (full) {
#pragma unroll
        for (int i = 0; i < 8; ++i)
            hp[(size_t)i * DIM] = c[i] + bv;          // row rowBase + i + 8*half
    } else {
#pragma unroll
        for (int i = 0; i < 8; ++i) {
            int row = rowBase + i + 8 * half;
            if (row < nNodes) hp[(size_t)i * DIM] = c[i] + bv;
        }
    }
}

// --------------------------- edge aggregation ------------------------------
// One wave per edge: 32 lanes x float4 = 128 columns. h-row gather is a 512B
// contiguous read (L2-resident). Scatter = native no-return f32 atomics,
// one VGPR address pair + immediate offsets.
__global__ void __launch_bounds__(256) gnn_aggregate(const float* __restrict__ h,
                                                     const int*   __restrict__ rows,
                                                     const int*   __restrict__ cols,
                                                     const float* __restrict__ vals,
                                                     float* __restrict__ out,
                                                     int nEdges) {
    const int edge = blockIdx.x * (blockDim.x >> 5) + (threadIdx.x >> 5);
    if (edge >= nEdges) return;
    const int lane = threadIdx.x & 31;

    const int   r  = rows[edge];
    const int   cc = cols[edge];
    const float v  = vals[edge];

    const float4 m = ((const float4*)(h + (size_t)cc * DIM))[lane];

    const uint64_t addr = (uint64_t)(uintptr_t)(out + (size_t)r * DIM + lane * 4);
    asm volatile(
        "global_atomic_add_f32 %0, %1, off\n\t"
        "global_atomic_add_f32 %0, %2, off offset:4\n\t"
        "global_atomic_add_f32 %0, %3, off offset:8\n\t"
        "global_atomic_add_f32 %0, %4, off offset:12"
        :: "v"(addr), "v"(v * m.x), "v"(v * m.y), "v"(v * m.z), "v"(v * m.w)
        : "memory");
}

// ------------------------------- launcher ----------------------------------
extern "C" void kernel_launch(void* const* d_in, const int* in_sizes, int n_in,
                              void* d_out, int out_size, void* d_ws, size_t ws_size,
                              hipStream_t stream) {
    const float* x    = (const float*)d_in[0];   // [N, 128]
    const float* w    = (const float*)d_in[1];   // [128, 128]
    const float* bias = (const float*)d_in[2];   // [128]
    const int*   rows = (const int*)  d_in[3];   // [E]
    const int*   cols = (const int*)  d_in[4];   // [E]
    const float* vals = (const float*)d_in[5];   // [E]
    float*       out  = (float*)d_out;

    const int nNodes = in_sizes[0] / DIM;
    const int nEdges = in_sizes[3];

    // Workspace: h [nNodes*128] floats, then Wt [128*128] floats (~51.3MB).
    float* h  = (float*)d_ws;
    float* wt = h + (size_t)nNodes * DIM;

    {   // 1) zero the poisoned output
        int blocks = (out_size + 255) / 256;
        gnn_zero_out<<<blocks, 256, 0, stream>>>(out, out_size);
    }
    {   // 2) transpose W for coalesced B-fragment loads
        int blocks = (DIM * DIM + 255) / 256;
        gnn_transpose_w<<<blocks, 256, 0, stream>>>(w, wt);
    }
    {   // 3) fp32 WMMA GEMM + bias -> h
        int blocks = (nNodes + 15) / 16;
        gnn_gemm_bias<<<blocks, 256, 0, stream>>>(x, wt, bias, h, nNodes);
    }
    {   // 4) wave-per-edge scatter-aggregate -> out
        int wavesPerBlock = 256 / 32;
        int blocks = (nEdges + wavesPerBlock - 1) / wavesPerBlock;
        gnn_aggregate<<<blocks, 256, 0, stream>>>(h, rows, cols, vals, out, nEdges);
    }
}